// ParallelCausalConv1d_37357625541009
// MI455X (gfx1250) — compile-verified
//
#include <hip/hip_runtime.h>
#include <stdint.h>
#include <stddef.h>

// Causal depthwise conv1d, B=8 L=4096 D=2048 K=4, fp32.
// Memory-bound (AI ~ 1 flop/byte): target is streaming 537 MB at HBM rate.
// Uses CDNA5 async global->LDS B128 DMA (ASYNCcnt) double-buffer pipeline,
// register sliding window for the K-1 tap history, NT b128 stores.

typedef float v4f __attribute__((ext_vector_type(4)));

constexpr int B_ = 8;
constexpr int L_ = 4096;
constexpr int D_ = 2048;

constexpr int THREADS = 128;               // 4 waves per block
constexpr int CPT     = 4;                 // channels per thread (float4)
constexpr int CPB     = THREADS * CPT;     // 512 channels per block
constexpr int TC      = 8;                 // time rows per chunk
constexpr int NCHUNK  = 16;                // chunks per block
constexpr int TB      = TC * NCHUNK;       // 128 rows per block

__global__ __launch_bounds__(THREADS)
void causal_dwconv4_kernel(const float* __restrict__ x,
                           const float* __restrict__ kern,
                           const float* __restrict__ bias,
                           float* __restrict__ out)
{
    // Double-buffered staging tile: 2 * 8 * 128 * 16B = 32 KB LDS.
    __shared__ v4f lds[2][TC][THREADS];

    const int tid = threadIdx.x;
    const int c   = blockIdx.x * CPB + tid * CPT;   // channel of this thread
    const int t0  = blockIdx.y * TB;                // first time row of block
    const int b   = blockIdx.z;

    const float* gcol = x   + ((size_t)b * L_) * D_ + c;
    float*       ocol = out + ((size_t)b * L_) * D_ + c;

    // Raw LDS byte offset of this thread's slot (low 32 bits of the
    // generic pointer to a __shared__ object are the DS address).
    const uint32_t lds_base = (uint32_t)(uintptr_t)(&lds[0][0][0]);
    const uint32_t lane_off = (uint32_t)(tid * (int)sizeof(v4f));
    constexpr uint32_t ROW_B = (uint32_t)(THREADS * sizeof(v4f)); // 2048 B
    constexpr uint32_t BUF_B = (uint32_t)(TC * THREADS * sizeof(v4f));

    // Issue one chunk (TC rows) of async global->LDS B128 copies.
    // Each lane moves its own float4 per row; ASYNCcnt += TC per call.
    auto issue_chunk = [&](int buf, int row0) {
        const float* g0 = gcol + (size_t)row0 * D_;
        uint32_t l0 = lds_base + (uint32_t)buf * BUF_B + lane_off;
#pragma unroll
        for (int r = 0; r < TC; ++r) {
            uint32_t       lo = l0 + (uint32_t)r * ROW_B;
            const float*   ga = g0 + (size_t)r * D_;
            asm volatile("global_load_async_to_lds_b128 %0, %1, off"
                         :: "v"(lo), "v"(ga)
                         : "memory");
        }
    };

    // Per-channel taps and bias (tiny, L2-resident across blocks).
    const v4f w0 = *(const v4f*)(kern + 0 * D_ + c);
    const v4f w1 = *(const v4f*)(kern + 1 * D_ + c);
    const v4f w2 = *(const v4f*)(kern + 2 * D_ + c);
    const v4f w3 = *(const v4f*)(kern + 3 * D_ + c);
    const v4f bv = *(const v4f*)(bias + c);

    // Prime the pipeline: chunks 0 and 1 in flight (ASYNCcnt = 2*TC).
    issue_chunk(0, t0);
    issue_chunk(1, t0 + TC);

    // Sliding-window history x[t-1], x[t-2], x[t-3] in registers.
    const v4f zero = {0.f, 0.f, 0.f, 0.f};
    v4f xm1 = (t0 - 1 >= 0) ? *(const v4f*)(gcol + (size_t)(t0 - 1) * D_) : zero;
    v4f xm2 = (t0 - 2 >= 0) ? *(const v4f*)(gcol + (size_t)(t0 - 2) * D_) : zero;
    v4f xm3 = (t0 - 3 >= 0) ? *(const v4f*)(gcol + (size_t)(t0 - 3) * D_) : zero;

    int t = t0;
#pragma unroll 1
    for (int ch = 0; ch < NCHUNK; ++ch) {
        // Wait until chunk `ch` has landed in LDS. While computing it,
        // chunk ch+1 (TC ops) stays in flight.
        if (ch + 1 < NCHUNK) {
            asm volatile("s_wait_asynccnt 8" ::: "memory");
        } else {
            asm volatile("s_wait_asynccnt 0" ::: "memory");
        }

#pragma unroll
        for (int r = 0; r < TC; ++r) {
            v4f x0 = lds[ch & 1][r][tid];
            v4f o  = bv;
            o += w3 * x0;    // tap k=3 -> x[t]
            o += w2 * xm1;   // tap k=2 -> x[t-1]
            o += w1 * xm2;   // tap k=1 -> x[t-2]
            o += w0 * xm3;   // tap k=0 -> x[t-3]
            __builtin_nontemporal_store(o, (v4f*)(ocol + (size_t)(t + r) * D_));
            xm3 = xm2; xm2 = xm1; xm1 = x0;
        }

        // WAR insurance: our LDS reads of this buffer have retired before
        // we hand the buffer back to the DMA engine.
        asm volatile("s_wait_dscnt 0" ::: "memory");
        if (ch + 2 < NCHUNK) {
            issue_chunk(ch & 1, t0 + (ch + 2) * TC);
        }
        t += TC;
    }
}

extern "C" void kernel_launch(void* const* d_in, const int* in_sizes, int n_in,
                              void* d_out, int out_size, void* d_ws, size_t ws_size,
                              hipStream_t stream)
{
    (void)in_sizes; (void)n_in; (void)out_size; (void)d_ws; (void)ws_size;
    const float* x    = (const float*)d_in[0];
    const float* kern = (const float*)d_in[1];
    const float* bias = (const float*)d_in[2];
    float*       out  = (float*)d_out;

    dim3 grid(D_ / CPB, L_ / TB, B_);   // 4 x 32 x 8 = 1024 blocks
    causal_dwconv4_kernel<<<grid, THREADS, 0, stream>>>(x, kern, bias, out);
}